// DifferentiableRocket_40269613367472
// MI455X (gfx1250) — compile-verified
//
#include <hip/hip_runtime.h>
#include <hip/hip_bf16.h>
#include <math.h>

// CDNA5 (gfx1250) fused ROCKET kernel:
//   conv(x[64,1,2048], kernels[2000,1,9]) -> max/mean over positions -> linear head.
// Conv is expressed as a GEMM  kernels[2000x9(pad12)] @ patches(x)[9x2040]
// using V_WMMA_F32_16X16X4_F32 (full fp32 precision, matches reference).

typedef __attribute__((ext_vector_type(2))) float v2f;
typedef __attribute__((ext_vector_type(8))) float v8f;

#define B_BATCH 64
#define LEN     2048
#define NKER    2000
#define KTAP    9
#define LOUT    2040            // 2048 - 9 + 1
#define NCLS    10
#define NFEAT   (2 * NKER)      // 4000, interleaved [max, mean]

__device__ __forceinline__ v8f wmma_f32_step(v2f a, v2f b, v8f c) {
  // 8 args: (neg_a, A, neg_b, B, c_mod, C, reuse_a, reuse_b)
  return __builtin_amdgcn_wmma_f32_16x16x4_f32(
      false, a, false, b, (short)0, c, false, false);
}

__global__ __launch_bounds__(32)
void rocket_conv_wmma(const float* __restrict__ x,        // [64, 2048]
                      const float* __restrict__ kernels,  // [2000, 9]
                      float* __restrict__ feats)          // [64, 4000]
{
  const int lane = threadIdx.x;        // 0..31 (wave32)
  const int col  = lane & 15;          // N column within tile
  const int half = lane >> 4;          // 0: lanes 0-15, 1: lanes 16-31
  const int kt   = blockIdx.x;         // kernel tile, 0..124
  const int bb   = blockIdx.y;         // batch, 0..63

  const float* xr = x + (size_t)bb * LEN;

  // ---- Load A (16 filters x 12 taps, zero padded past K=8) ----
  // A 16x4 layout: lane l holds M=l%16; VGPR0 = K(4s + 2*half), VGPR1 = K+1.
  const float* kp = kernels + (size_t)(kt * 16 + col) * KTAP;
  const int k0 = half * 2;             // 0 or 2
  v2f a0, a1, a2;
  a0.x = kp[k0];      a0.y = kp[k0 + 1];
  a1.x = kp[4 + k0];  a1.y = kp[4 + k0 + 1];
  a2.x = (half == 0) ? kp[8] : 0.0f;   // K=8 valid only; K=9..11 are padding
  a2.y = 0.0f;

  v8f sumAcc = {};
  v8f maxAcc;
#pragma unroll
  for (int v = 0; v < 8; ++v) maxAcc[v] = -INFINITY;

  // B 4x16 layout: lane l supplies rows K = 4s + 2*half (+0,+1) at column col.
  const int base = col + k0;

  // ---- 127 full 16-column chunks (positions 0..2031) ----
  int n0 = 0;
  for (int chunk = 0; chunk < 127; ++chunk, n0 += 16) {
    const float* p = xr + n0 + base;   // max index: 2016+17+9 = 2042 < 2048
    v2f b0, b1, b2;
    b0.x = p[0]; b0.y = p[1];
    b1.x = p[4]; b1.y = p[5];
    b2.x = p[8]; b2.y = p[9];

    v8f c = {};
    c = wmma_f32_step(a0, b0, c);
    c = wmma_f32_step(a1, b1, c);
    c = wmma_f32_step(a2, b2, c);

#pragma unroll
    for (int v = 0; v < 8; ++v) {
      sumAcc[v] += c[v];
      maxAcc[v]  = fmaxf(maxAcc[v], c[v]);
    }
  }

  // ---- Last partial chunk: n0 = 2032, valid columns 0..7 ----
  {
    float vb[6];
#pragma unroll
    for (int i = 0; i < 6; ++i) {
      const int offs = (i >> 1) * 4 + (i & 1);     // 0,1,4,5,8,9
      const int idx  = n0 + base + offs;
      vb[i] = (idx < LEN) ? xr[idx] : 0.0f;        // clamp (pollutes only invalid cols)
    }
    v2f b0, b1, b2;
    b0.x = vb[0]; b0.y = vb[1];
    b1.x = vb[2]; b1.y = vb[3];
    b2.x = vb[4]; b2.y = vb[5];

    v8f c = {};
    c = wmma_f32_step(a0, b0, c);
    c = wmma_f32_step(a1, b1, c);
    c = wmma_f32_step(a2, b2, c);

    const bool valid = (n0 + col) < LOUT;          // cols 0..7
#pragma unroll
    for (int v = 0; v < 8; ++v) {
      sumAcc[v] += valid ? c[v] : 0.0f;
      maxAcc[v]  = valid ? fmaxf(maxAcc[v], c[v]) : maxAcc[v];
    }
  }

  // ---- Reduce the 16 columns within each half-wave, write features ----
  const float invL = 1.0f / (float)LOUT;
#pragma unroll
  for (int v = 0; v < 8; ++v) {
    float s = sumAcc[v];
    float m = maxAcc[v];
#pragma unroll
    for (int off = 1; off < 16; off <<= 1) {       // stays inside 16-lane halves
      s += __shfl_xor(s, off, 32);
      m  = fmaxf(m, __shfl_xor(m, off, 32));
    }
    if (col == 0) {                                // lanes 0 and 16
      const int row = kt * 16 + v + 8 * half;      // global filter index
      float* fp = feats + (size_t)bb * NFEAT + 2 * row;
      fp[0] = m;                                   // max
      fp[1] = s * invL;                            // mean
    }
  }
}

// ---- Tiny linear head: out[b,c] = feats[b,:] . W[c,:] + bias[c] ----
__global__ __launch_bounds__(32)
void rocket_head(const float* __restrict__ feats,  // [64, 4000]
                 const float* __restrict__ W,      // [10, 4000]
                 const float* __restrict__ bias,   // [10]
                 float* __restrict__ out)          // [64, 10]
{
  const int lane = threadIdx.x;
  const int c = blockIdx.x % NCLS;
  const int b = blockIdx.x / NCLS;
  const float* f = feats + (size_t)b * NFEAT;
  const float* w = W + (size_t)c * NFEAT;
  float acc = 0.0f;
  for (int i = lane; i < NFEAT; i += 32) acc = fmaf(f[i], w[i], acc);
#pragma unroll
  for (int off = 16; off > 0; off >>= 1) acc += __shfl_xor(acc, off, 32);
  if (lane == 0) out[b * NCLS + c] = acc + bias[c];
}

extern "C" void kernel_launch(void* const* d_in, const int* in_sizes, int n_in,
                              void* d_out, int out_size, void* d_ws, size_t ws_size,
                              hipStream_t stream) {
  const float* x       = (const float*)d_in[0];   // [64,1,2048]
  const float* kernels = (const float*)d_in[1];   // [2000,1,9]
  const float* W       = (const float*)d_in[2];   // [10,4000]
  const float* bias    = (const float*)d_in[3];   // [10]
  float* out   = (float*)d_out;                   // [64,10]
  float* feats = (float*)d_ws;                    // [64,4000] scratch (1.0 MB)

  dim3 grid1(NKER / 16, B_BATCH);                 // 125 x 64 waves
  rocket_conv_wmma<<<grid1, dim3(32), 0, stream>>>(x, kernels, feats);

  rocket_head<<<dim3(B_BATCH * NCLS), dim3(32), 0, stream>>>(feats, W, bias, out);
}